// FeatureExtractorModule_52776558133915
// MI455X (gfx1250) — compile-verified
//
#include <hip/hip_runtime.h>
#include <hip/hip_bf16.h>
#include <cstddef>

// ---------------------------------------------------------------------------
// DynamicEdgeConv x2 (DGCNN feature extractor) for MI455X (gfx1250, wave32).
//
// Edge MLP first layer factored: hidden(i,j) = relu(u_i + v_j) with
//   u = x @ (W1_top - W1_bot) + b1,  v = x @ W1_bot
// so per-edge work collapses to gather+add. All heavy GEMMs (64-dim Gram for
// kNN, u/v projections, per-edge hidden @ W2) run on V_WMMA_F32_16X16X4_F32.
// ---------------------------------------------------------------------------

typedef __attribute__((ext_vector_type(2))) float v2f;
typedef __attribute__((ext_vector_type(8))) float v8f;

#define KNN_K  10
#define FDIM   64
#define NPB    4096            // points per batch
#define NBATCH 8
#define NPTS   (NPB * NBATCH)  // 32768

// f32 WMMA 16x16x4, 8-arg form: (neg_a, A, neg_b, B, c_mod, C, reuse_a, reuse_b)
__device__ __forceinline__ v8f wmma_f32_k4(v2f a, v2f b, v8f c) {
  return __builtin_amdgcn_wmma_f32_16x16x4_f32(false, a, false, b, (short)0, c,
                                               false, false);
}

// ---------------------------------------------------------------------------
// Squared norms per point (C = 3 or 64).
// ---------------------------------------------------------------------------
__global__ void sqnorm_kernel(const float* __restrict__ src,
                              float* __restrict__ sq, int npts, int c) {
  int p = blockIdx.x * blockDim.x + threadIdx.x;
  if (p >= npts) return;
  float s = 0.f;
  for (int i = 0; i < c; ++i) {
    float vv = src[(size_t)p * c + i];
    s += vv * vv;
  }
  sq[p] = s;
}

// ---------------------------------------------------------------------------
// Brute-force kNN for C=3 (layer-1 graph). One thread per query point; LDS
// tiles of 512 candidate points; top-10 kept in registers via an unrolled
// sorted insert (constant indices -> stays in VGPRs, no scratch spill).
// ---------------------------------------------------------------------------
__global__ __launch_bounds__(256) void knn3_kernel(
    const float* __restrict__ x, const float* __restrict__ sqx,
    int* __restrict__ idx_out) {
  __shared__ float sx[512], sy[512], sz[512], ss[512];
  const int b = blockIdx.x >> 4;            // 16 WGs per batch
  const int r0 = (blockIdx.x & 15) * 256;
  const int t = threadIdx.x;
  const int p = b * NPB + r0 + t;

  const float xi = x[p * 3 + 0], yi = x[p * 3 + 1], zi = x[p * 3 + 2];
  const float si = sqx[p];

  float td[KNN_K];
  int ti[KNN_K];
#pragma unroll
  for (int j = 0; j < KNN_K; ++j) { td[j] = 3.4e38f; ti[j] = p; }

  for (int cb = 0; cb < NPB; cb += 512) {
    for (int j = t; j < 512; j += 256) {
      int q = b * NPB + cb + j;
      sx[j] = x[q * 3 + 0];
      sy[j] = x[q * 3 + 1];
      sz[j] = x[q * 3 + 2];
      ss[j] = sqx[q];
    }
    __syncthreads();
    for (int j = 0; j < 512; ++j) {
      float d = si + ss[j] - 2.f * (xi * sx[j] + yi * sy[j] + zi * sz[j]);
      if (d < td[KNN_K - 1]) {
        td[KNN_K - 1] = d;
        ti[KNN_K - 1] = b * NPB + cb + j;
#pragma unroll
        for (int s = KNN_K - 1; s > 0; --s) {
          if (td[s] < td[s - 1]) {
            float tf = td[s]; td[s] = td[s - 1]; td[s - 1] = tf;
            int   tt = ti[s]; ti[s] = ti[s - 1]; ti[s - 1] = tt;
          }
        }
      }
    }
    __syncthreads();
  }
#pragma unroll
  for (int j = 0; j < KNN_K; ++j) idx_out[(size_t)p * KNN_K + j] = ti[j];
}

// ---------------------------------------------------------------------------
// Layer-1 u/v (K = 3, too small for WMMA): one thread per (point, channel).
// ---------------------------------------------------------------------------
__global__ void uv3_kernel(const float* __restrict__ x,
                           const float* __restrict__ W1,
                           const float* __restrict__ b1,
                           float* __restrict__ u, float* __restrict__ v) {
  int f = blockIdx.x * blockDim.x + threadIdx.x;
  if (f >= NPTS * FDIM) return;
  int p = f >> 6, h = f & 63;
  float x0 = x[p * 3 + 0], x1 = x[p * 3 + 1], x2 = x[p * 3 + 2];
  float wt0 = W1[0 * 64 + h], wt1 = W1[1 * 64 + h], wt2 = W1[2 * 64 + h];
  float wb0 = W1[3 * 64 + h], wb1 = W1[4 * 64 + h], wb2 = W1[5 * 64 + h];
  v[f] = x0 * wb0 + x1 * wb1 + x2 * wb2;
  u[f] = x0 * (wt0 - wb0) + x1 * (wt1 - wb1) + x2 * (wt2 - wb2) + b1[h];
}

// ---------------------------------------------------------------------------
// Wd = W1b[0:64] - W1b[64:128];  Wb = W1b[64:128]   (both 64x64 row-major)
// ---------------------------------------------------------------------------
__global__ void prepw_kernel(const float* __restrict__ W1b,
                             float* __restrict__ Wd, float* __restrict__ Wb) {
  int f = blockIdx.x * blockDim.x + threadIdx.x;
  if (f >= 64 * 64) return;
  float top = W1b[f];
  float bot = W1b[64 * 64 + f];
  Wd[f] = top - bot;
  Wb[f] = bot;
}

// ---------------------------------------------------------------------------
// Layer-2 u/v: dense (NPTS x 64) @ (64 x 64) GEMMs on WMMA.
// WG = 256 thr = 8 waves = 128 rows; each wave keeps its 16x64 A panel in
// 32 VGPRs and reuses it for both weight matrices (u and v) x 4 col tiles.
// ---------------------------------------------------------------------------
__global__ __launch_bounds__(256) void uv64_kernel(
    const float* __restrict__ A, const float* __restrict__ Wd,
    const float* __restrict__ Wb, const float* __restrict__ b1,
    float* __restrict__ u, float* __restrict__ v) {
  __shared__ float sWd[64 * 64];
  __shared__ float sWb[64 * 64];
  const int t = threadIdx.x, wave = t >> 5, lane = t & 31;
  const int half = lane >> 4, ln = lane & 15;

  for (int f = t; f < 64 * 64; f += 256) { sWd[f] = Wd[f]; sWb[f] = Wb[f]; }

  const int r0 = blockIdx.x * 128 + wave * 16;
  float aA[32];
  {
    const float* arow = A + (size_t)(r0 + ln) * FDIM;
#pragma unroll
    for (int kk = 0; kk < 16; ++kk) {
      aA[2 * kk + 0] = arow[4 * kk + 2 * half + 0];
      aA[2 * kk + 1] = arow[4 * kk + 2 * half + 1];
    }
  }
  __syncthreads();

  for (int which = 0; which < 2; ++which) {
    const float* W = which ? sWb : sWd;
    float* out = which ? v : u;
    for (int nt = 0; nt < 4; ++nt) {
      v8f acc = {};
#pragma unroll
      for (int kk = 0; kk < 16; ++kk) {
        v2f a;  a.x = aA[2 * kk];  a.y = aA[2 * kk + 1];
        v2f bb;
        bb.x = W[(4 * kk + 2 * half + 0) * 64 + nt * 16 + ln];
        bb.y = W[(4 * kk + 2 * half + 1) * 64 + nt * 16 + ln];
        acc = wmma_f32_k4(a, bb, acc);
      }
      float bias = which ? 0.f : b1[nt * 16 + ln];
#pragma unroll
      for (int vv = 0; vv < 8; ++vv) {
        int row = r0 + vv + 8 * half;
        out[(size_t)row * FDIM + nt * 16 + ln] = acc[vv] + bias;
      }
    }
  }
}

// ---------------------------------------------------------------------------
// Fused edge MLP: out_i = max_j relu(u_i + v_{idx[i][j]}) @ W2 + b2.
// WG handles 16 points = 160 edges, split into two 80-edge passes so all LDS
// (hid 80x66 pad, W2 64x64, res 80x64) stays < 64 KB static.
// GEMM: 20 tiles/pass of 16 edges x 16 cols, 16 K-steps each, on WMMA.
// ---------------------------------------------------------------------------
#define EMLP_PTS 16
#define HPAD 66   // LDS row stride for hid: 66 % 64 == 2 -> conflict-free col reads

__global__ __launch_bounds__(256) void edge_mlp_kernel(
    const float* __restrict__ u, const float* __restrict__ v,
    const int* __restrict__ idx, const float* __restrict__ W2,
    const float* __restrict__ b2, float* __restrict__ out) {
  __shared__ float hid[80 * HPAD];   // relu(u_i + v_j) for 80 edges
  __shared__ float sW2[64 * 64];
  __shared__ float res[80 * 64];     // hid @ W2
  const int t = threadIdx.x, wave = t >> 5, lane = t & 31;
  const int half = lane >> 4, ln = lane & 15;
  const int p0 = blockIdx.x * EMLP_PTS;

  for (int f = t; f < 64 * 64; f += 256) sW2[f] = W2[f];

  for (int pass = 0; pass < 2; ++pass) {
    // phase 1: build hidden activations for this pass's 80 edges
    for (int f = t; f < 80 * FDIM; f += 256) {
      int el = f >> 6, h = f & 63;
      int e = pass * 80 + el;
      int p = p0 + e / KNN_K;
      int nb = idx[(size_t)p * KNN_K + (e % KNN_K)];
      float hv = u[(size_t)p * FDIM + h] + v[(size_t)nb * FDIM + h];
      hid[el * HPAD + h] = hv > 0.f ? hv : 0.f;
    }
    __syncthreads();

    // phase 2: (80 x 64) @ (64 x 64) via 5x4 WMMA tiles spread over 8 waves
    for (int tile = wave; tile < 20; tile += 8) {
      int et = tile >> 2, nt = tile & 3;
      v8f acc = {};
#pragma unroll
      for (int kk = 0; kk < 16; ++kk) {
        v2f a;
        a.x = hid[(et * 16 + ln) * HPAD + 4 * kk + 2 * half + 0];
        a.y = hid[(et * 16 + ln) * HPAD + 4 * kk + 2 * half + 1];
        v2f bb;
        bb.x = sW2[(4 * kk + 2 * half + 0) * 64 + nt * 16 + ln];
        bb.y = sW2[(4 * kk + 2 * half + 1) * 64 + nt * 16 + ln];
        acc = wmma_f32_k4(a, bb, acc);
      }
#pragma unroll
      for (int vv = 0; vv < 8; ++vv)
        res[(et * 16 + vv + 8 * half) * 64 + nt * 16 + ln] = acc[vv];
    }
    __syncthreads();

    // phase 3: max over the 10 edges of each of this pass's 8 points
    for (int f = t; f < 8 * FDIM; f += 256) {
      int pl = f >> 6, h = f & 63;
      float m = -3.4e38f;
#pragma unroll
      for (int j = 0; j < KNN_K; ++j) {
        float rv = res[(pl * KNN_K + j) * 64 + h];
        m = m > rv ? m : rv;
      }
      out[(size_t)(p0 + pass * 8 + pl) * FDIM + h] = m + b2[h];
    }
    __syncthreads();
  }
}

// ---------------------------------------------------------------------------
// kNN over 64-dim features: Gram matrix on WMMA + fully parallel top-10.
// WG = 16 query rows; 8 waves each own a 16-col slice of a 128-col chunk.
// A panel (16 rows x 64 feats) lives in 32 VGPRs for the whole 4096-col sweep.
// Top-k: each of 256 threads keeps a register top-10 over its (row, slice)
// stripe (8 candidates/chunk); the 16 sorted partial lists per row are merged
// once at the end (early-exit on sorted slices).
// ---------------------------------------------------------------------------
__global__ __launch_bounds__(256) void knn64_kernel(
    const float* __restrict__ feat, const float* __restrict__ sqf,
    int* __restrict__ idx_out) {
  __shared__ float dist[16 * 128];
  __shared__ float sqr[16];
  __shared__ float md[16 * 16 * KNN_K];
  __shared__ int   mi[16 * 16 * KNN_K];
  const int b = blockIdx.x >> 8;              // 256 row-blocks per batch
  const int r0 = (blockIdx.x & 255) * 16;
  const int t = threadIdx.x, wave = t >> 5, lane = t & 31;
  const int half = lane >> 4, ln = lane & 15;
  const int mrow = t >> 4;                    // merge: row owned by this thread
  const int msub = t & 15;                    // merge: 8-wide column slice

  if (t < 16) sqr[t] = sqf[b * NPB + r0 + t];

  float aA[32];
  {
    const float* arow = feat + (size_t)(b * NPB + r0 + ln) * FDIM;
#pragma unroll
    for (int kk = 0; kk < 16; ++kk) {
      aA[2 * kk + 0] = arow[4 * kk + 2 * half + 0];
      aA[2 * kk + 1] = arow[4 * kk + 2 * half + 1];
    }
  }

  float td[KNN_K];
  int ti[KNN_K];
#pragma unroll
  for (int j = 0; j < KNN_K; ++j) { td[j] = 3.4e38f; ti[j] = 0; }

  __syncthreads();

  for (int cb = 0; cb < NPB; cb += 128) {
    const int colp = b * NPB + cb + wave * 16 + ln;
    const float* brow = feat + (size_t)colp * FDIM;
    const float sqc = sqf[colp];

    // keep the next chunk's B panel warm (global_prefetch_b8)
    if (cb + 128 < NPB) __builtin_prefetch(brow + 128 * FDIM, 0, 1);

    v8f acc = {};
#pragma unroll
    for (int kk = 0; kk < 16; ++kk) {
      v2f a;  a.x = aA[2 * kk];  a.y = aA[2 * kk + 1];
      v2f bb;
      bb.x = brow[4 * kk + 2 * half + 0];
      bb.y = brow[4 * kk + 2 * half + 1];
      acc = wmma_f32_k4(a, bb, acc);
    }
#pragma unroll
    for (int vv = 0; vv < 8; ++vv) {
      int lr = vv + 8 * half;
      dist[lr * 128 + wave * 16 + ln] = sqr[lr] + sqc - 2.f * acc[vv];
    }
    __syncthreads();

    // parallel partial top-k: all 256 threads, 8 candidates each
#pragma unroll
    for (int j = 0; j < 8; ++j) {
      int c = msub * 8 + j;
      float d = dist[mrow * 128 + c];
      if (d < td[KNN_K - 1]) {
        td[KNN_K - 1] = d;
        ti[KNN_K - 1] = b * NPB + cb + c;
#pragma unroll
        for (int s = KNN_K - 1; s > 0; --s) {
          if (td[s] < td[s - 1]) {
            float tf = td[s]; td[s] = td[s - 1]; td[s - 1] = tf;
            int   tt = ti[s]; ti[s] = ti[s - 1]; ti[s - 1] = tt;
          }
        }
      }
    }
    __syncthreads();
  }

  // publish sorted partial lists
#pragma unroll
  for (int j = 0; j < KNN_K; ++j) {
    md[(mrow * 16 + msub) * KNN_K + j] = td[j];
    mi[(mrow * 16 + msub) * KNN_K + j] = ti[j];
  }
  __syncthreads();

  // final merge: one thread per row, 16 sorted lists, early-exit per list
  if (t < 16) {
    float fd[KNN_K];
    int fi[KNN_K];
#pragma unroll
    for (int j = 0; j < KNN_K; ++j) { fd[j] = 3.4e38f; fi[j] = 0; }
    for (int s = 0; s < 16; ++s) {
      for (int j = 0; j < KNN_K; ++j) {
        float d = md[(t * 16 + s) * KNN_K + j];
        if (d >= fd[KNN_K - 1]) break;      // rest of this sorted list is larger
        fd[KNN_K - 1] = d;
        fi[KNN_K - 1] = mi[(t * 16 + s) * KNN_K + j];
#pragma unroll
        for (int s2 = KNN_K - 1; s2 > 0; --s2) {
          if (fd[s2] < fd[s2 - 1]) {
            float tf = fd[s2]; fd[s2] = fd[s2 - 1]; fd[s2 - 1] = tf;
            int   tt = fi[s2]; fi[s2] = fi[s2 - 1]; fi[s2 - 1] = tt;
          }
        }
      }
    }
    int p = b * NPB + r0 + t;
#pragma unroll
    for (int j = 0; j < KNN_K; ++j) idx_out[(size_t)p * KNN_K + j] = fi[j];
  }
}

// ---------------------------------------------------------------------------
// Host-side launch. Inputs (setup_inputs order):
//  0:x [BN*3] 1:batch(unused) 2:W1a[6*64] 3:b1a[64] 4:W2a[64*64] 5:b2a[64]
//  6:W1b[128*64] 7:b1b[64] 8:W2b[64*64] 9:b2b[64]
// Output: h2 [BN*64] float.
// Workspace layout (~28 MB, assumed <= ws_size).
// ---------------------------------------------------------------------------
extern "C" void kernel_launch(void* const* d_in, const int* in_sizes, int n_in,
                              void* d_out, int out_size, void* d_ws,
                              size_t ws_size, hipStream_t stream) {
  (void)in_sizes; (void)n_in; (void)out_size; (void)ws_size;

  const float* x   = (const float*)d_in[0];
  const float* W1a = (const float*)d_in[2];
  const float* b1a = (const float*)d_in[3];
  const float* W2a = (const float*)d_in[4];
  const float* b2a = (const float*)d_in[5];
  const float* W1b = (const float*)d_in[6];
  const float* b1b = (const float*)d_in[7];
  const float* W2b = (const float*)d_in[8];
  const float* b2b = (const float*)d_in[9];

  float* ws  = (float*)d_ws;
  float* h1  = ws;                               // NPTS*64
  float* uu  = h1 + (size_t)NPTS * FDIM;         // NPTS*64
  float* vv  = uu + (size_t)NPTS * FDIM;         // NPTS*64
  float* sqx = vv + (size_t)NPTS * FDIM;         // NPTS
  float* sqh = sqx + NPTS;                       // NPTS
  float* Wd  = sqh + NPTS;                       // 64*64
  float* Wb  = Wd + 64 * 64;                     // 64*64
  int*   idx1 = (int*)(Wb + 64 * 64);            // NPTS*10
  int*   idx2 = idx1 + (size_t)NPTS * KNN_K;     // NPTS*10

  // ---- layer 1 ----
  sqnorm_kernel<<<(NPTS + 255) / 256, 256, 0, stream>>>(x, sqx, NPTS, 3);
  knn3_kernel<<<NBATCH * (NPB / 256), 256, 0, stream>>>(x, sqx, idx1);
  uv3_kernel<<<(NPTS * FDIM + 255) / 256, 256, 0, stream>>>(x, W1a, b1a, uu, vv);
  edge_mlp_kernel<<<NPTS / EMLP_PTS, 256, 0, stream>>>(uu, vv, idx1, W2a, b2a, h1);

  // ---- layer 2 ----
  sqnorm_kernel<<<(NPTS + 255) / 256, 256, 0, stream>>>(h1, sqh, NPTS, FDIM);
  knn64_kernel<<<NPTS / 16, 256, 0, stream>>>(h1, sqh, idx2);
  prepw_kernel<<<(64 * 64 + 255) / 256, 256, 0, stream>>>(W1b, Wd, Wb);
  uv64_kernel<<<NPTS / 128, 256, 0, stream>>>(h1, Wd, Wb, b1b, uu, vv);
  edge_mlp_kernel<<<NPTS / EMLP_PTS, 256, 0, stream>>>(uu, vv, idx2, W2b, b2b,
                                                       (float*)d_out);
}